// RealNVP2D_8100308320936
// MI455X (gfx1250) — compile-verified
//
#include <hip/hip_runtime.h>
#include <hip/hip_bf16.h>

#define LAYERS  8
#define HD      64
#define NREG    (LAYERS * 2)          // (layer, net) regions
#define RSTRIDE 208                   // floats per region: r[64], (slope,icept)[65], pad
#define LUTSZ   (NREG * RSTRIDE)      // 3328 floats = 13312 bytes (multiple of 16)

// ---------- fast transcendentals (gfx1250 TRANS ops where available) ----------
__device__ __forceinline__ float fast_exp2(float x) {
#if __has_builtin(__builtin_amdgcn_exp2f)
    return __builtin_amdgcn_exp2f(x);          // v_exp_f32
#else
    return exp2f(x);
#endif
}
__device__ __forceinline__ float fast_exp(float x) {
    return fast_exp2(x * 1.44269504088896340736f);
}
__device__ __forceinline__ float fast_rcp(float x) {
#if __has_builtin(__builtin_amdgcn_rcpf)
    return __builtin_amdgcn_rcpf(x);           // v_rcp_f32
#else
    return 1.0f / x;
#endif
}
__device__ __forceinline__ float fast_tanh(float x) {
#if __has_builtin(__builtin_amdgcn_tanhf)
    return __builtin_amdgcn_tanhf(x);          // v_tanh_f32 (CDNA5 TRANS)
#else
    float e = fast_exp(2.0f * x);              // tanh = 1 - 2/(exp(2x)+1)
    return 1.0f - 2.0f * fast_rcp(e + 1.0f);
#endif
}

// ---------- prep: MLP -> sorted-breakpoint piecewise-linear tables ----------
// y(z) = A z + C + sum_j c_j |z - r_j|,  c_j = 0.495 w2_j |w1_j|, r_j = -b1_j/w1_j
// Segment k (k = #{r_j <= z}, r sorted):  y = slope_k z + icept_k
//   slope_k = A + 2*Sc_k - Sc_64 ;  icept_k = C + Scr_64 - 2*Scr_k
// All loops unroll-disabled: runs once on 16 threads; keep code size tiny.
__global__ void realnvp_prep(const float* __restrict__ sw1, const float* __restrict__ sb1,
                             const float* __restrict__ sw2, const float* __restrict__ sb2,
                             const float* __restrict__ tw1, const float* __restrict__ tb1,
                             const float* __restrict__ tw2, const float* __restrict__ tb2,
                             float* __restrict__ P) {
    int t = threadIdx.x;                       // 0..15 -> (layer, net)
    if (t >= NREG) return;
    int l = t >> 1, n = t & 1;
    const float* w1 = (n ? tw1 : sw1) + l * HD;
    const float* b1 = (n ? tb1 : sb1) + l * HD;
    const float* w2 = (n ? tw2 : sw2) + l * HD;
    float        bb = (n ? tb2 : sb2)[l];

    float c[HD], r[HD];
    float A = 0.0f, C = bb;
#pragma unroll 1
    for (int j = 0; j < HD; ++j) {
        float w1j = w1[j], b1j = b1[j], w2j = w2[j];
        A = fmaf(0.505f * w2j, w1j, A);
        C = fmaf(0.505f * w2j, b1j, C);
        if (w1j != 0.0f) {
            c[j] = 0.495f * w2j * fabsf(w1j);
            r[j] = -b1j / w1j;
        } else {                               // constant contribution
            c[j] = 0.0f; r[j] = 3.0e38f;       // sorts last, contributes nothing
            C = fmaf(0.495f * w2j, fabsf(b1j), C);
        }
    }
    // insertion sort by r
#pragma unroll 1
    for (int j = 1; j < HD; ++j) {
        float rk = r[j], ck = c[j];
        int m = j - 1;
#pragma unroll 1
        while (m >= 0 && r[m] > rk) { r[m + 1] = r[m]; c[m + 1] = c[m]; --m; }
        r[m + 1] = rk; c[m + 1] = ck;
    }
    float Sc64 = 0.0f, Scr64 = 0.0f;
#pragma unroll 1
    for (int j = 0; j < HD; ++j) { Sc64 += c[j]; Scr64 += c[j] * r[j]; }

    float* p = P + t * RSTRIDE;
#pragma unroll 1
    for (int j = 0; j < HD; ++j) p[j] = r[j];
    float Sc = 0.0f, Scr = 0.0f;
#pragma unroll 1
    for (int k = 0; k <= HD; ++k) {
        p[64 + 2 * k]     = A + 2.0f * Sc - Sc64;     // slope_k
        p[64 + 2 * k + 1] = C + Scr64 - 2.0f * Scr;   // icept_k
        if (k < HD) { Sc += c[k]; Scr += c[k] * r[k]; }
    }
#pragma unroll 1
    for (int j = 64 + 130; j < RSTRIDE; ++j) p[j] = 0.0f;   // deterministic pad
}

// ---------- main: async-stage LUT to LDS, branchless binary search per layer ----------
__global__ __launch_bounds__(512) void realnvp_main(const float2* __restrict__ z,
                                                    const float* __restrict__ P,
                                                    float* __restrict__ out, int nsamp) {
    __shared__ float lut[LUTSZ];
    {   // CDNA5 async global->LDS DMA (b128 per lane), tracked by ASYNCcnt
        unsigned ldsbase = (unsigned)(uintptr_t)(&lut[0]);   // generic low 32 = LDS offset
        unsigned long long gbase = (unsigned long long)(uintptr_t)P;
        for (int idx = threadIdx.x; idx < LUTSZ / 4; idx += 512) {
            unsigned           laddr = ldsbase + (unsigned)idx * 16u;
            unsigned long long gaddr = gbase + (unsigned long long)idx * 16ull;
            asm volatile("global_load_async_to_lds_b128 %0, %1, off"
                         :: "v"(laddr), "v"(gaddr) : "memory");
        }
        asm volatile("s_wait_asynccnt 0x0" ::: "memory");
    }
    __syncthreads();

    int i = blockIdx.x * 512 + threadIdx.x;
    if (i >= nsamp) return;
    float2 zz = z[i];
    float a = zz.x, b = zz.y, ld = 0.0f;
    const char* Lb = (const char*)lut;         // LDS-rooted byte pointer
#pragma unroll
    for (int l = 0; l < LAYERS; ++l) {
        const int SOFF = (2 * l) * RSTRIDE * 4;          // compile-time region offsets
        const int TOFF = SOFF + RSTRIDE * 4;             // (fit 16-bit DS immediates)
        // branchless lower_bound over sorted r[64]; cursor kept in BYTES so each
        // probe is ds_load_b32 [kb], offset:(region + (st-1)*4)
        int kbs = 0, kbt = 0;
#pragma unroll
        for (int st = 32; st >= 1; st >>= 1) {
            float rs = *(const float*)(Lb + SOFF + (kbs + (st - 1) * 4));
            float rt = *(const float*)(Lb + TOFF + (kbt + (st - 1) * 4));
            kbs += (rs <= a) ? st * 4 : 0;
            kbt += (rt <= a) ? st * 4 : 0;
        }
        kbs += (*(const float*)(Lb + SOFF + kbs) <= a) ? 4 : 0;   // k in [0,64]
        kbt += (*(const float*)(Lb + TOFF + kbt) <= a) ? 4 : 0;
        // segment coeffs: pairs start at +256 B; pair k at +256 + 8k = +256 + 2*kb
        float2 ps = *(const float2*)(Lb + SOFF + 256 + 2 * kbs);  // ds_load_b64
        float2 pt = *(const float2*)(Lb + TOFF + 256 + 2 * kbt);
        float s  = 3.0f * fast_tanh(fmaf(ps.x, a, ps.y));
        float t  = fmaf(pt.x, a, pt.y);
        float bn = fmaf(b, fast_exp(s), t);
        b = a; a = bn;
        ld += s;
    }
    ((float2*)out)[i]  = make_float2(a, b);
    out[2 * nsamp + i] = ld;
}

// ---------- fallback (workspace too small): direct leaky-ReLU form ----------
__global__ __launch_bounds__(256) void realnvp_direct(const float2* __restrict__ z,
        const float* __restrict__ sw1, const float* __restrict__ sb1,
        const float* __restrict__ sw2, const float* __restrict__ sb2,
        const float* __restrict__ tw1, const float* __restrict__ tb1,
        const float* __restrict__ tw2, const float* __restrict__ tb2,
        float* __restrict__ out, int nsamp) {
    int i = blockIdx.x * 256 + threadIdx.x;
    if (i >= nsamp) return;
    float2 zz = z[i];
    float a = zz.x, b = zz.y, ld = 0.0f;
#pragma unroll
    for (int l = 0; l < LAYERS; ++l) {
        const float* ws1 = sw1 + l * HD; const float* wsb = sb1 + l * HD;
        const float* ws2 = sw2 + l * HD;
        const float* wt1 = tw1 + l * HD; const float* wtb = tb1 + l * HD;
        const float* wt2 = tw2 + l * HD;
        float accs = sb2[l], acct = tb2[l];
#pragma unroll 4
        for (int j = 0; j < HD; ++j) {
            float hs = fmaf(a, ws1[j], wsb[j]);
            hs = fmaxf(hs, 0.01f * hs);
            accs = fmaf(hs, ws2[j], accs);
            float ht = fmaf(a, wt1[j], wtb[j]);
            ht = fmaxf(ht, 0.01f * ht);
            acct = fmaf(ht, wt2[j], acct);
        }
        float s  = 3.0f * fast_tanh(accs);
        float bn = fmaf(b, fast_exp(s), acct);
        b = a; a = bn;
        ld += s;
    }
    out[2 * i] = a; out[2 * i + 1] = b;
    out[2 * nsamp + i] = ld;
}

extern "C" void kernel_launch(void* const* d_in, const int* in_sizes, int n_in,
                              void* d_out, int out_size, void* d_ws, size_t ws_size,
                              hipStream_t stream) {
    const float* z   = (const float*)d_in[0];
    const float* sw1 = (const float*)d_in[1];
    const float* sb1 = (const float*)d_in[2];
    const float* sw2 = (const float*)d_in[3];
    const float* sb2 = (const float*)d_in[4];
    const float* tw1 = (const float*)d_in[5];
    const float* tb1 = (const float*)d_in[6];
    const float* tw2 = (const float*)d_in[7];
    const float* tb2 = (const float*)d_in[8];
    float* out = (float*)d_out;

    int nsamp = in_sizes[0] / 2;
    size_t need = (size_t)LUTSZ * sizeof(float);

    if (d_ws != nullptr && ws_size >= need) {
        float* P = (float*)d_ws;
        realnvp_prep<<<1, 16, 0, stream>>>(sw1, sb1, sw2, sb2, tw1, tb1, tw2, tb2, P);
        int blocks = (nsamp + 511) / 512;
        realnvp_main<<<blocks, 512, 0, stream>>>((const float2*)z, P, out, nsamp);
    } else {
        int blocks = (nsamp + 255) / 256;
        realnvp_direct<<<blocks, 256, 0, stream>>>((const float2*)z,
            sw1, sb1, sw2, sb2, tw1, tb1, tw2, tb2, out, nsamp);
    }
}